// BERT_CRF_NER_31430570672625
// MI455X (gfx1250) — compile-verified
//
#include <hip/hip_runtime.h>
#include <hip/hip_bf16.h>
#include <math.h>

typedef __attribute__((ext_vector_type(2))) float v2f;
typedef __attribute__((ext_vector_type(8))) float v8f;

#define B_      256
#define T_      256
#define H_      768
#define L_      13
#define LP      16          // labels padded to 16 for WMMA N dimension
#define START_  11
#define NEGV    (-10000.0f)
#define WSTRIDE 772         // LDS row stride (dwords); 772 % 64 == 4 -> conflict-free b64 reads

// -----------------------------------------------------------------------------
// Kernel 1: feats[r][n] = sum_h X[r][h] * W[n][h] + b[n]   (r < 65536, n < 16)
// One wave = one 16-row M tile. WMMA fp32 16x16x4, K looped 768/4 = 192 times.
// fp32 A-frag layout (ISA 7.12.2): lane l holds A[m][k+2*half .. +1] (contig b64)
// fp32 B-frag layout (4x16, row-striped): lane l holds B[k+2*half+v][n=m]
//   = W[m][k+2*half+v]  -> same contiguous b64 addressing, served from LDS.
// -----------------------------------------------------------------------------
__global__ __launch_bounds__(256) void feats_wmma_kernel(
    const float* __restrict__ X, const float* __restrict__ W,
    const float* __restrict__ bias, float* __restrict__ feats)
{
  __shared__ float ldsW[16 * WSTRIDE];

  const int tid = threadIdx.x;
  // Stage W padded to 16 rows (rows 13..15 zero) — coalesced, one-time.
  for (int i = tid; i < 16 * H_; i += 256) {
    int r = i / H_, c = i - r * H_;
    ldsW[r * WSTRIDE + c] = (r < L_) ? W[r * H_ + c] : 0.0f;
  }
  __syncthreads();

  const int wave = tid >> 5;
  const int lane = tid & 31;
  const int half = lane >> 4;   // 0: lanes 0-15, 1: lanes 16-31
  const int m    = lane & 15;

  const int rowbase = (blockIdx.x * 8 + wave) * 16;
  const float* arow = X + (size_t)(rowbase + m) * H_ + 2 * half;
  const float* brow = ldsW + m * WSTRIDE + 2 * half;

  v8f acc = {};
  #pragma unroll 8
  for (int k = 0; k < H_; k += 4) {
    v2f a = *(const v2f*)(arow + k);   // global_load_b64, streams each row once
    v2f b = *(const v2f*)(brow + k);   // ds_load_b64, bank-conflict-free
    acc = __builtin_amdgcn_wmma_f32_16x16x4_f32(
        /*neg_a=*/false, a, /*neg_b=*/false, b,
        /*c_mod=*/(short)0, acc, /*reuse_a=*/false, /*reuse_b=*/false);
  }

  // C/D layout: VGPR r holds D[row = half*8 + r][n = m]
  const float bv = (m < L_) ? bias[m] : 0.0f;
  float* outrow = feats + (size_t)(rowbase + half * 8) * LP + m;
  #pragma unroll
  for (int r = 0; r < 8; ++r) {
    outrow[(size_t)r * LP] = acc[r] + bv;
  }
}

// -----------------------------------------------------------------------------
// Kernel 2: Viterbi decode. One wave per block; lanes 0-15 = batch item 2*bid,
// lanes 16-31 = item 2*bid+1; lane's (l & 15) is the "to" label.
// Backpointers kept in LDS so the serial backtrack never touches HBM.
// -----------------------------------------------------------------------------
__global__ __launch_bounds__(32) void viterbi_kernel(
    const float* __restrict__ feats, const float* __restrict__ trans,
    float* __restrict__ out)
{
  __shared__ unsigned char psi[(T_ - 1) * 32];

  const int  lane     = threadIdx.x;
  const int  halfBase = lane & 16;
  const int  to       = lane & 15;
  const int  item     = blockIdx.x * 2 + (lane >> 4);
  const bool active   = (to < L_);

  float tr[L_];
  #pragma unroll
  for (int f = 0; f < L_; ++f)
    tr[f] = active ? trans[to * L_ + f] : NEGV;

  float ld = (to == START_) ? 0.0f : NEGV;   // ld0; feats[t=0] is never added (matches ref)
  const float* frow = feats + (size_t)item * T_ * LP + to;

  for (int t = 1; t < T_; ++t) {
    float best = -3.4e38f;
    int   bp   = 0;
    #pragma unroll
    for (int f = 0; f < L_; ++f) {
      float v = tr[f] + __shfl(ld, halfBase | f, 32);
      if (v > best) { best = v; bp = f; }   // strict > == first-index argmax
    }
    psi[(t - 1) * 32 + halfBase + to] = (unsigned char)bp;
    float ft = frow[(size_t)t * LP];
    ld = active ? (best + ft) : NEGV;
  }
  __syncthreads();

  // max_p = max(softmax(ld)) = 1 / sum(exp(ld - ld_max)); last = argmax(ld)
  float mx = -3.4e38f;
  int   am = 0;
  #pragma unroll
  for (int f = 0; f < L_; ++f) {
    float v = __shfl(ld, halfBase | f, 32);
    if (v > mx) { mx = v; am = f; }
  }
  float s = 0.0f;
  #pragma unroll
  for (int f = 0; f < L_; ++f) {
    float v = __shfl(ld, halfBase | f, 32);
    s += __expf(v - mx);
  }
  float maxp = 1.0f / s;

  if (to == 0) {
    out[item] = maxp * (1.0f / (float)T_);
    float* pathOut = out + B_ + (size_t)item * T_;
    int cur = am;
    pathOut[T_ - 1] = (float)cur;
    for (int t = T_ - 2; t >= 0; --t) {
      cur = psi[t * 32 + halfBase + cur];
      pathOut[t] = (float)cur;
    }
  }
}

// -----------------------------------------------------------------------------
// Launch: d_in = {bert_seq_out[256,256,768] f32, W[13,768] f32, b[13] f32,
//                 transitions[13,13] f32}; d_out = 256 max_p + 256*256 path (f32).
// d_ws: feats padded [65536][16] f32 = 4 MB.
// -----------------------------------------------------------------------------
extern "C" void kernel_launch(void* const* d_in, const int* in_sizes, int n_in,
                              void* d_out, int out_size, void* d_ws, size_t ws_size,
                              hipStream_t stream) {
  (void)in_sizes; (void)n_in; (void)out_size; (void)ws_size;
  const float* X     = (const float*)d_in[0];
  const float* W     = (const float*)d_in[1];
  const float* bias  = (const float*)d_in[2];
  const float* trans = (const float*)d_in[3];
  float* out   = (float*)d_out;
  float* feats = (float*)d_ws;   // 65536 * 16 floats

  // 65536 rows / (8 waves * 16 rows) = 512 blocks
  feats_wmma_kernel<<<512, 256, 0, stream>>>(X, W, bias, feats);
  // 256 batch items / 2 per wave = 128 blocks
  viterbi_kernel<<<128, 32, 0, stream>>>(feats, trans, out);
}